// CRF_layer_14147622273702
// MI455X (gfx1250) — compile-verified
//
#include <hip/hip_runtime.h>
#include <hip/hip_bf16.h>

// ---------------------------------------------------------------------------
// CRF layer for gfx1250 (MI455X).
//   T=256, B=128, D=1024, K=21 (padded to 32)
// Phase 1: emissions = feat(32768x1024) @ W(1024x21) + b  via V_WMMA_F32_16X16X4_F32
//          (memory-bound: 134 MB activation stream @ 23.3 TB/s ~ 5.8 us)
// Phase 2: viterbi + CRF-NLL recursions, one wave32 per (batch, mode);
//          score broadcast via v_readlane (__shfl with constant lane) --
//          no barriers / DS traffic in the 255-step critical path.
// ---------------------------------------------------------------------------

#define T_DIM 256
#define B_DIM 128
#define D_DIM 1024
#define K_TAGS 21
#define KP 32                      // padded tag dimension
#define M_ROWS (T_DIM * B_DIM)     // 32768
#define NEG_BIG (-1.0e30f)

typedef float v2f __attribute__((ext_vector_type(2)));
typedef float v8f __attribute__((ext_vector_type(8)));

// ---------------------------------------------------------------------------
// Kernel 0: pack W (D x 21, row major) into the exact WMMA B-operand lane
// layout.  Flat index -> (s, ntile, lane, pair):
//   k   = 4*s + 2*(lane>>4) + pair
//   col = ntile*16 + (lane&15)
// so in the GEMM each lane does one contiguous 8B load per WMMA step.
// ---------------------------------------------------------------------------
__global__ __launch_bounds__(256) void crf_pack_w(const float* __restrict__ W,
                                                  float* __restrict__ Wpk) {
    int idx   = blockIdx.x * 256 + threadIdx.x;     // 0 .. 32767
    int p     = idx & 1;
    int l     = (idx >> 1) & 31;
    int ntile = (idx >> 6) & 1;
    int s     = idx >> 7;                           // 0 .. 255
    int k     = 4 * s + 2 * (l >> 4) + p;
    int col   = ntile * 16 + (l & 15);
    float v = 0.0f;
    if (col < K_TAGS) v = W[k * K_TAGS + col];
    Wpk[idx] = v;
}

// ---------------------------------------------------------------------------
// Kernel 1: WMMA GEMM.  One wave per 16-row M tile, two 16-col N tiles.
// 256 blocks x 256 threads (8 waves) -> 2048 M tiles.
// ---------------------------------------------------------------------------
__global__ __launch_bounds__(256) void crf_gemm(const float* __restrict__ A,
                                                const float* __restrict__ Wpk,
                                                const float* __restrict__ bias,
                                                float* __restrict__ E) {
    const int lane  = threadIdx.x & 31;
    const int wave  = threadIdx.x >> 5;
    const int mtile = blockIdx.x * 8 + wave;        // 0 .. 2047

    const int row0  = lane & 15;                    // A row within tile
    const int khalf = lane >> 4;                    // 0: K+0/K+1, 1: K+2/K+3

    const float* arow = A + (size_t)(mtile * 16 + row0) * D_DIM + 2 * khalf;

    v8f c0 = {};                                    // cols 0..15
    v8f c1 = {};                                    // cols 16..31 (21.. zero-padded)

#pragma unroll 8
    for (int s = 0; s < D_DIM / 4; ++s) {
        v2f a  = *(const v2f*)(arow + 4 * s);
        v2f b0 = *(const v2f*)(Wpk + s * 128 + lane * 2);
        v2f b1 = *(const v2f*)(Wpk + s * 128 + 64 + lane * 2);
        c0 = __builtin_amdgcn_wmma_f32_16x16x4_f32(false, a, false, b0,
                                                   (short)0, c0, false, false);
        c1 = __builtin_amdgcn_wmma_f32_16x16x4_f32(false, a, false, b1,
                                                   (short)0, c1, false, false);
    }

    // Epilogue: C layout -> lane holds col (lane&15), rows khalf*8 + v.
    const int col0 = lane & 15;
    const int col1 = 16 + (lane & 15);
    const float bia0 = bias[col0];                          // col0 < 16 < 21
    const float bia1 = (col1 < K_TAGS) ? bias[col1] : 0.0f; // pad cols -> 0

#pragma unroll
    for (int v = 0; v < 8; ++v) {
        const size_t r = (size_t)(mtile * 16 + khalf * 8 + v);
        E[r * KP + col0] = c0[v] + bia0;
        E[r * KP + col1] = c1[v] + bia1;
    }
}

// ---------------------------------------------------------------------------
// Kernel 2: CRF recursions.  grid = (B, 2), block = 32 (one wave).
//   blockIdx.y == 0 : viterbi max-product + numerator score + backtrace
//   blockIdx.y == 1 : logsumexp forward (partition function denominator)
// Lane j owns next-state j; trans[:,j] cached in 21 VGPRs; previous score
// vector broadcast via __shfl with constant lane index (v_readlane_b32),
// so the 255-step sequential loop has no barriers and no LDS round-trips.
// Back-pointers live in LDS (1 byte store/lane/step), read at backtrace.
// ---------------------------------------------------------------------------
__global__ __launch_bounds__(32) void crf_recur(const float* __restrict__ E,
                                                const int* __restrict__ tags,
                                                const float* __restrict__ startT,
                                                const float* __restrict__ endT,
                                                const float* __restrict__ trans,
                                                float* __restrict__ out,
                                                float* __restrict__ num_ws,
                                                float* __restrict__ den_ws) {
    __shared__ float sv[32];
    __shared__ unsigned char hist[T_DIM * 32];

    const int j = threadIdx.x;          // state (0..20 active, 21..31 padding)
    const int b = blockIdx.x;           // batch element
    const bool vmode = (blockIdx.y == 0);
    const bool act = (j < K_TAGS);

    float tc[K_TAGS];                   // trans[i][j] for all prev-states i
#pragma unroll
    for (int i = 0; i < K_TAGS; ++i) tc[i] = act ? trans[i * K_TAGS + j] : 0.0f;

    float score = act ? (startT[j] + E[(size_t)b * KP + j]) : NEG_BIG;

    // lane-0 scalars for the numerator path (viterbi mode)
    int   ptag = tags[b];
    float num  = startT[ptag] + E[(size_t)b * KP + ptag];

    for (int t = 1; t < T_DIM; ++t) {
        const int   tg = tags[t * B_DIM + b];
        const bool  mk = (tg != 0);
        const float emit = E[((size_t)t * B_DIM + b) * KP + j];

        if (vmode) {
            // ---- viterbi: max / argmax over prev-states (first-max ties) ----
            float best = NEG_BIG;
            int   bid  = 0;
#pragma unroll
            for (int i = 0; i < K_TAGS; ++i) {
                float c = __shfl(score, i, 32) + tc[i];   // v_readlane broadcast
                if (c > best) { best = c; bid = i; }
            }
            hist[t * 32 + j] = (unsigned char)(mk ? bid : j);
            if (mk && act) score = best + emit;
            if (j == 0 && mk) {
                num += trans[ptag * K_TAGS + tg] +
                       E[((size_t)t * B_DIM + b) * KP + tg];
                ptag = tg;
            }
        } else {
            // ---- forward logsumexp over prev-states ----
            float cc[K_TAGS];
            float m = NEG_BIG;
#pragma unroll
            for (int i = 0; i < K_TAGS; ++i) {
                cc[i] = __shfl(score, i, 32) + tc[i];     // v_readlane broadcast
                m = fmaxf(m, cc[i]);
            }
            float s = 0.0f;
#pragma unroll
            for (int i = 0; i < K_TAGS; ++i) s += __expf(cc[i] - m);
            if (mk && act) score = m + __logf(s) + emit;
        }
    }

    // -------- termination --------
    sv[j] = act ? (score + endT[j]) : NEG_BIG;
    __syncthreads();                      // also makes hist[] visible to lane 0

    if (j == 0) {
        if (vmode) {
            float best = NEG_BIG;
            int   bt   = 0;
            for (int i = 0; i < K_TAGS; ++i)
                if (sv[i] > best) { best = sv[i]; bt = i; }
            num_ws[b] = num + endT[ptag];

            // backtrace through LDS back-pointers
            int cur = bt;
            for (int t = T_DIM - 1; t >= 1; --t) {
                const int tg = tags[t * B_DIM + b];
                out[t * B_DIM + b] = (tg != 0) ? (float)cur : 0.0f;
                cur = (int)hist[t * 32 + cur];
            }
            out[b] = (tags[b] != 0) ? (float)cur : 0.0f;
        } else {
            float m = NEG_BIG;
            for (int i = 0; i < K_TAGS; ++i) m = fmaxf(m, sv[i]);
            float s = 0.0f;
            for (int i = 0; i < K_TAGS; ++i) s += __expf(sv[i] - m);
            den_ws[b] = m + __logf(s);
        }
    }
}

// ---------------------------------------------------------------------------
// Kernel 3: deterministic fixed-order loss reduction.
// ---------------------------------------------------------------------------
__global__ void crf_loss(const float* __restrict__ num_ws,
                         const float* __restrict__ den_ws,
                         float* __restrict__ out) {
    if (threadIdx.x == 0 && blockIdx.x == 0) {
        float loss = 0.0f;
        for (int b = 0; b < B_DIM; ++b) loss += den_ws[b] - num_ws[b];
        out[M_ROWS] = loss;   // -sum(num - den)
    }
}

// ---------------------------------------------------------------------------
extern "C" void kernel_launch(void* const* d_in, const int* in_sizes, int n_in,
                              void* d_out, int out_size, void* d_ws, size_t ws_size,
                              hipStream_t stream) {
    const float* feat   = (const float*)d_in[0];   // (T,B,D) f32
    const int*   tags   = (const int*)d_in[1];     // (T,B)   i32
    const float* W      = (const float*)d_in[2];   // (D,K)   f32
    const float* bias   = (const float*)d_in[3];   // (K,)
    const float* startT = (const float*)d_in[4];   // (K,)
    const float* endT   = (const float*)d_in[5];   // (K,)
    const float* trans  = (const float*)d_in[6];   // (K,K)

    float* out = (float*)d_out;                    // [T*B path floats][1 loss]
    float* ws  = (float*)d_ws;

    float* E      = ws;                            // 32768*32 f32   (4 MB)
    float* Wpk    = E + (size_t)M_ROWS * KP;       // 32768 f32     (128 KB)
    float* num_ws = Wpk + 32768;                   // 128 f32
    float* den_ws = num_ws + B_DIM;                // 128 f32

    crf_pack_w<<<128, 256, 0, stream>>>(W, Wpk);
    crf_gemm<<<256, 256, 0, stream>>>(feat, Wpk, bias, E);
    crf_recur<<<dim3(B_DIM, 2), 32, 0, stream>>>(E, tags, startT, endT, trans,
                                                 out, num_ws, den_ws);
    crf_loss<<<1, 32, 0, stream>>>(num_ws, den_ws, out);
}